// BoundaryModulePlus_82136954569042
// MI455X (gfx1250) — compile-verified
//
#include <hip/hip_runtime.h>

// ---------------------------------------------------------------------------
// Model dims (fixed by the reference)
// ---------------------------------------------------------------------------
#define NB    2
#define CINC  128
#define TSEQ  256
#define NSMP  10
#define NDUR  40
#define D0    256
#define D1    128
#define NHEADS 4
#define DHD   32                 // head dim
#define CH    40                 // conv H (= DUR)
#define CW    256                // conv W (= T)
#define CHW   (CH*CW)            // 10240
#define PHW   ((CH/2)*(CW/2))    // 2560 (after 2x2 pool)
#define SMPC  (NSMP*NDUR*TSEQ)   // 102400 columns of smp_weight

typedef __attribute__((ext_vector_type(2))) float v2f;
typedef __attribute__((ext_vector_type(8))) float v8f;

__device__ __forceinline__ v8f wmma4(v2f a, v2f b, v8f c) {
  // V_WMMA_F32_16X16X4_F32 : D = A(16x4) x B(4x16) + C(16x16), fp32
  return __builtin_amdgcn_wmma_f32_16x16x4_f32(false, a, false, b, (short)0, c,
                                               false, false);
}

__device__ __forceinline__ float activate(float v, int act) {
  if (act == 1) return v > 0.f ? v : 0.01f * v;          // leaky
  if (act == 2) return v > 0.f ? v : 0.f;                // relu
  if (act == 3) return 1.f / (1.f + __expf(-v));         // sigmoid
  return v;
}

// ---------------------------------------------------------------------------
// Generic strided/batched WMMA GEMM:  C = epilogue(alpha*(A@B + biasN + biasM))
// A: [M,K] row-major (lda), batch offsets z1*sA1 + z2*sA2
// BTR=0: B[K,N] row-major (ldb). BTR=1: element (k,n) = Bt[n*ldb + k].
// NT = 16x16 tiles per wave along N (A fragment reused NT times per K-step).
// M%16==0, K%4==0 required; NT=4 requires N%64==0.
// Block: 128 threads = 4 waves.
// ---------------------------------------------------------------------------
struct GemmP {
  const float* A;  int lda; long sA1; long sA2;
  const float* Bm; int ldb; long sB1; long sB2;
  float* C;        int ldc; long sC1; long sC2;
  const float* biasN; const float* biasM;
  const float* scaleM; const float* shiftM;
  const float* resid;           // indexed exactly like C
  float alpha; int act;
  int M, N, K, bz2;
};

template <int BTR, int NT>
__global__ __launch_bounds__(128) void wmma_gemm(GemmP p) {
  const int wave = threadIdx.x >> 5;
  const int lane = threadIdx.x & 31;
  const int kh   = lane >> 4;
  const int lm   = lane & 15;
  const int nbase = (blockIdx.x * 4 + wave) * (16 * NT);
  if (nbase >= p.N) return;        // wave-uniform exit (EXEC stays full)
  const int m0 = blockIdx.y * 16;
  const int z1 = blockIdx.z / p.bz2, z2 = blockIdx.z % p.bz2;
  const float* __restrict__ A  = p.A  + z1 * p.sA1 + z2 * p.sA2;
  const float* __restrict__ Bm = p.Bm + z1 * p.sB1 + z2 * p.sB2;
  float* __restrict__       C  = p.C  + z1 * p.sC1 + z2 * p.sC2;

  const float* Arow = A + (long)(m0 + lm) * p.lda;   // A-frag row
  v8f acc[NT];
#pragma unroll
  for (int j = 0; j < NT; ++j) acc[j] = (v8f){};

  for (int k0 = 0; k0 < p.K; k0 += 4) {
    const int k = k0 + 2 * kh;
    v2f a;
    a.x = Arow[k];
    a.y = Arow[k + 1];
#pragma unroll
    for (int j = 0; j < NT; ++j) {
      const int n = nbase + j * 16 + lm;
      v2f b;
      if (BTR) {
        b.x = Bm[(long)n * p.ldb + k];
        b.y = Bm[(long)n * p.ldb + k + 1];
      } else {
        b.x = Bm[(long)k * p.ldb + n];
        b.y = Bm[(long)(k + 1) * p.ldb + n];
      }
      acc[j] = wmma4(a, b, acc[j]);
    }
  }

#pragma unroll
  for (int j = 0; j < NT; ++j) {
    const int n = nbase + j * 16 + lm;
    const float bn_ = p.biasN ? p.biasN[n] : 0.f;
#pragma unroll
    for (int g = 0; g < 8; ++g) {
      const int mm = m0 + g + 8 * kh;
      float v = acc[j][g] + bn_ + (p.biasM ? p.biasM[mm] : 0.f);
      v *= p.alpha;
      if (p.scaleM) v = v * p.scaleM[mm] + p.shiftM[mm];
      if (p.resid)
        v += p.resid[z1 * p.sC1 + z2 * p.sC2 + (long)mm * p.ldc + n];
      C[(long)mm * p.ldc + n] = activate(v, p.act);
    }
  }
}

// ---------------------------------------------------------------------------
// Fused BM layer + block0 (loop-interchanged: A fragment of x reused across
// all 10 n-tiles; LDS B fragment reused across both o-tiles in phase 2):
//   r[c, n, t16] = X[b](128x256) @ smp columns      -> LDS (128x160)
//   y0[o, t16]   = leaky(w0[o, c*10+n] @ r + b0[o])
// grid = (T/16, DUR, B), 256 threads (8 waves), dyn LDS = 128*160*4 = 80KB
// ---------------------------------------------------------------------------
__global__ __launch_bounds__(256) void bm_block0(
    const float* __restrict__ x, const float* __restrict__ smp,
    const float* __restrict__ w0, const float* __restrict__ b0,
    float* __restrict__ y0) {
  extern __shared__ float rblk[];                 // [128][160]
  const int wave = threadIdx.x >> 5, lane = threadIdx.x & 31;
  const int kh = lane >> 4, lm = lane & 15;
  const int t0 = blockIdx.x * 16;
  const int d  = blockIdx.y;
  const int b  = blockIdx.z;

  // phase 1: each wave owns c-rows [wave*16, wave*16+16), all 10 n-tiles
  {
    const int c0 = wave * 16;
    const float* xrow = x + (long)b * CINC * TSEQ + (long)(c0 + lm) * TSEQ;
    const float* scol = smp + (long)d * TSEQ + t0 + lm;  // + k*SMPC + n*10240
    v8f acc[NSMP];
#pragma unroll
    for (int jt = 0; jt < NSMP; ++jt) acc[jt] = (v8f){};
    for (int k0 = 0; k0 < TSEQ; k0 += 4) {
      const int k = k0 + 2 * kh;
      v2f a;
      a.x = xrow[k];
      a.y = xrow[k + 1];
      const float* s0 = scol + (long)k * SMPC;
#pragma unroll
      for (int jt = 0; jt < NSMP; ++jt) {
        v2f bb;
        bb.x = s0[jt * (NDUR * TSEQ)];
        bb.y = s0[jt * (NDUR * TSEQ) + SMPC];
        acc[jt] = wmma4(a, bb, acc[jt]);
      }
    }
#pragma unroll
    for (int jt = 0; jt < NSMP; ++jt)
#pragma unroll
      for (int g = 0; g < 8; ++g)
        rblk[(c0 + g + 8 * kh) * 160 + jt * 16 + lm] = acc[jt][g];
  }
  __syncthreads();

  // phase 2: each wave owns o-tiles wave*16 and wave*16+128
  {
    const float* w0a = w0 + (long)(wave * 16 + lm) * (CINC * NSMP);
    const float* w0b = w0a + 128L * (CINC * NSMP);
    v8f acc0 = {}, acc1 = {};
    for (int k0 = 0; k0 < CINC * NSMP; k0 += 4) {
      const int k = k0 + 2 * kh;
      const int c1 = k / NSMP, n1 = k - c1 * NSMP;
      const int k2 = k + 1;
      const int c2 = k2 / NSMP, n2 = k2 - c2 * NSMP;
      v2f bb;
      bb.x = rblk[c1 * 160 + n1 * 16 + lm];
      bb.y = rblk[c2 * 160 + n2 * 16 + lm];
      v2f a0, a1;
      a0.x = w0a[k];  a0.y = w0a[k + 1];
      a1.x = w0b[k];  a1.y = w0b[k + 1];
      acc0 = wmma4(a0, bb, acc0);
      acc1 = wmma4(a1, bb, acc1);
    }
    float* yb = y0 + (long)b * D0 * CHW + d * CW + t0 + lm;
#pragma unroll
    for (int g = 0; g < 8; ++g) {
      const int o0 = wave * 16 + g + 8 * kh;
      float v = acc0[g] + b0[o0];
      yb[(long)o0 * CHW] = v > 0.f ? v : 0.01f * v;
      const int o1 = o0 + 128;
      float u = acc1[g] + b0[o1];
      yb[(long)o1 * CHW] = u > 0.f ? u : 0.01f * u;
    }
  }
}

// ---------------------------------------------------------------------------
// Fused positional attention (flash-style; never materializes the 210MB map):
//   scores[16 q][2560 k] = th^T @ ph  -> LDS, softmax over k,
//   y[c, q] = sum_k g[c,k] * s[q,k]
// grid = (CHW/16, 1, B), 256 threads, dyn LDS = 16*2560*4 = 160KB
// ---------------------------------------------------------------------------
__global__ __launch_bounds__(256) void pos_attn(
    const float* __restrict__ th, const float* __restrict__ php,
    const float* __restrict__ gp, float* __restrict__ ypos) {
  extern __shared__ float sc[];                 // [16][2560]
  __shared__ float red[16][17];
  const int wave = threadIdx.x >> 5, lane = threadIdx.x & 31;
  const int kh = lane >> 4, lm = lane & 15;
  const int q0 = blockIdx.x * 16;
  const int b  = blockIdx.z;
  const float* thb = th  + (long)b * D1 * CHW + q0 + lm;
  const float* phb = php + (long)b * D1 * PHW;
  const float* gb  = gp  + (long)b * D1 * PHW;

  // phase 1: wave owns key tiles [wave*20, wave*20+20), blocked 4 at a time
  for (int g4 = 0; g4 < 5; ++g4) {
    const int base = (wave * 20 + g4 * 4) * 16;
    v8f acc[4];
#pragma unroll
    for (int j = 0; j < 4; ++j) acc[j] = (v8f){};
    for (int k0 = 0; k0 < D1; k0 += 4) {
      const int k = k0 + 2 * kh;
      v2f a;                                     // A[q, c] = th[c, q]
      a.x = thb[(long)k * CHW];
      a.y = thb[(long)(k + 1) * CHW];
      const float* pr = phb + (long)k * PHW + base + lm;
#pragma unroll
      for (int j = 0; j < 4; ++j) {
        v2f bb;                                  // B[c, kcol] = ph[c, kcol]
        bb.x = pr[j * 16];
        bb.y = pr[j * 16 + PHW];
        acc[j] = wmma4(a, bb, acc[j]);
      }
    }
#pragma unroll
    for (int j = 0; j < 4; ++j)
#pragma unroll
      for (int g = 0; g < 8; ++g)
        sc[(g + 8 * kh) * PHW + base + j * 16 + lm] = acc[j][g];
  }
  __syncthreads();

  // softmax over k: 16 rows x 16 threads each
  {
    const int row = threadIdx.x >> 4;
    const int jj  = threadIdx.x & 15;
    float mx = -3.4e38f;
    for (int k = jj; k < PHW; k += 16) mx = fmaxf(mx, sc[row * PHW + k]);
    red[row][jj] = mx;
    __syncthreads();
    mx = red[row][0];
#pragma unroll
    for (int i = 1; i < 16; ++i) mx = fmaxf(mx, red[row][i]);
    __syncthreads();
    float sm = 0.f;
    for (int k = jj; k < PHW; k += 16) {
      const float e = __expf(sc[row * PHW + k] - mx);
      sc[row * PHW + k] = e;
      sm += e;
    }
    red[row][jj] = sm;
    __syncthreads();
    sm = 0.f;
#pragma unroll
    for (int i = 0; i < 16; ++i) sm += red[row][i];
    const float inv = 1.f / sm;
    for (int k = jj; k < PHW; k += 16) sc[row * PHW + k] *= inv;
  }
  __syncthreads();

  // phase 2: y[c, q] = g[c,:] . s[q,:]  (each wave owns one 16-c tile)
  {
    const int c0 = wave * 16;
    const float* grow = gb + (long)(c0 + lm) * PHW;
    const float* srow = sc + lm * PHW;
    v8f acc = {};
    for (int k0 = 0; k0 < PHW; k0 += 4) {
      const int k = k0 + 2 * kh;
      v2f a;
      a.x = grow[k];
      a.y = grow[k + 1];
      v2f bb;                                    // B[k, q] = s[q, k]
      bb.x = srow[k];
      bb.y = srow[k + 1];
      acc = wmma4(a, bb, acc);
    }
#pragma unroll
    for (int g = 0; g < 8; ++g)
      ypos[(long)b * D1 * CHW + (long)(c0 + g + 8 * kh) * CHW + q0 + lm] =
          acc[g];
  }
}

// ---------------------------------------------------------------------------
// Direct 3x3 conv (pad 1), optional BN + activation.  in/out: [B, ch, 40, 256]
// ---------------------------------------------------------------------------
__global__ __launch_bounds__(256) void conv3x3(
    const float* __restrict__ in, const float* __restrict__ w,
    const float* __restrict__ bias, const float* __restrict__ s,
    const float* __restrict__ t, float* __restrict__ out, int co, int ci,
    int act) {
  const long total = (long)NB * co * CHW;
  const long idx = (long)blockIdx.x * 256 + threadIdx.x;
  if (idx >= total) return;
  const int wp = idx % CW;
  long r = idx / CW;
  const int hp = r % CH;  r /= CH;
  const int o  = r % co;
  const int b  = r / co;
  const float* ip = in + (long)b * ci * CHW;
  const float* wk = w + (long)o * ci * 9;
  float acc = 0.f;
  for (int c = 0; c < ci; ++c) {
    const float* irow = ip + (long)c * CHW;
    __builtin_prefetch(irow + CHW, 0, 1);   // next input plane -> prefetch
    const float* wr = wk + c * 9;
#pragma unroll
    for (int dh = -1; dh <= 1; ++dh) {
      const int hh = hp + dh;
      if (hh < 0 || hh >= CH) continue;
#pragma unroll
      for (int dw = -1; dw <= 1; ++dw) {
        const int ww = wp + dw;
        if (ww < 0 || ww >= CW) continue;
        acc += irow[hh * CW + ww] * wr[(dh + 1) * 3 + (dw + 1)];
      }
    }
  }
  acc += bias[o];
  if (s) acc = acc * s[o] + t[o];
  out[idx] = activate(acc, act);
}

// ---------------------------------------------------------------------------
// Small helpers
// ---------------------------------------------------------------------------
__global__ __launch_bounds__(256) void softmax_rows(float* d, int L) {
  __shared__ float red[256];
  float* r = d + (long)blockIdx.x * L;
  const int tid = threadIdx.x;
  float mx = -3.4e38f;
  for (int i = tid; i < L; i += 256) mx = fmaxf(mx, r[i]);
  red[tid] = mx; __syncthreads();
  for (int s = 128; s > 0; s >>= 1) {
    if (tid < s) red[tid] = fmaxf(red[tid], red[tid + s]);
    __syncthreads();
  }
  mx = red[0]; __syncthreads();
  float sm = 0.f;
  for (int i = tid; i < L; i += 256) {
    const float e = __expf(r[i] - mx);
    r[i] = e; sm += e;
  }
  red[tid] = sm; __syncthreads();
  for (int s = 128; s > 0; s >>= 1) {
    if (tid < s) red[tid] += red[tid + s];
    __syncthreads();
  }
  const float inv = 1.f / red[0];
  for (int i = tid; i < L; i += 256) r[i] *= inv;
}

__global__ __launch_bounds__(256) void transpose_ct(const float* __restrict__ in,
                                                    float* __restrict__ out) {
  // [B,C,T] -> [B,T,C]
  const int i = blockIdx.x * 256 + threadIdx.x;
  if (i >= NB * CINC * TSEQ) return;
  const int c = i % CINC;
  const int t = (i / CINC) % TSEQ;
  const int b = i / (CINC * TSEQ);
  out[i] = in[(long)b * CINC * TSEQ + (long)c * TSEQ + t];
}

__global__ __launch_bounds__(256) void gate_fuse(const float* __restrict__ gv,
                                                 const float* __restrict__ vg,
                                                 const float* __restrict__ v,
                                                 float* __restrict__ x) {
  // x[b,c,t] = gv[b,t,c]*vg[b,t,c] + (1-gv)*v[b,t,c]
  const int i = blockIdx.x * 256 + threadIdx.x;
  if (i >= NB * CINC * TSEQ) return;
  const int t = i % TSEQ;
  const int c = (i / TSEQ) % CINC;
  const int b = i / (TSEQ * CINC);
  const long j = (long)b * TSEQ * CINC + (long)t * CINC + c;
  const float g = gv[j];
  x[i] = g * vg[j] + (1.f - g) * v[j];
}

__global__ __launch_bounds__(256) void pool2k(const float* __restrict__ in,
                                              float* __restrict__ out) {
  // [B,128,40,256] -> [B,128,20,128]
  const int OH = CH / 2, OW = CW / 2;
  const int i = blockIdx.x * 256 + threadIdx.x;
  if (i >= NB * D1 * OH * OW) return;
  const int w = i % OW;
  const int h = (i / OW) % OH;
  const int c = (i / (OW * OH)) % D1;
  const int b = i / (OW * OH * D1);
  const float* p = in + ((long)(b * D1 + c) * CH + 2 * h) * CW + 2 * w;
  out[i] = fmaxf(fmaxf(p[0], p[1]), fmaxf(p[CW], p[CW + 1]));
}

__global__ __launch_bounds__(256) void chan_shuffle(const float* __restrict__ y,
                                                    float* __restrict__ out) {
  // out.flat[b, i] = y[b, i % 128, i / 128]   (permute-then-view quirk)
  const int per = D1 * CHW;
  const int i = blockIdx.x * 256 + threadIdx.x;
  if (i >= NB * per) return;
  const int b = i / per, j = i % per;
  out[i] = y[(long)b * per + (long)(j % D1) * CHW + (j / D1)];
}

__global__ __launch_bounds__(256) void add2(const float* a, const float* b,
                                            float* o, int n) {
  const int i = blockIdx.x * 256 + threadIdx.x;
  if (i < n) o[i] = a[i] + b[i];
}

__global__ __launch_bounds__(256) void combine3(const float* a, const float* b,
                                                const float* c, float* o,
                                                int n) {
  const int i = blockIdx.x * 256 + threadIdx.x;
  if (i < n) o[i] = (a[i] + b[i] + c[i]) * (1.f / 3.f);
}

// ---------------------------------------------------------------------------
// Host side
// ---------------------------------------------------------------------------
static void gemm(hipStream_t st, const float* A, int lda, long sA1, long sA2,
                 const float* Bm, int ldb, long sB1, long sB2, int btr,
                 float* C, int ldc, long sC1, long sC2, int M, int N, int K,
                 int z1, int z2, const float* bN, const float* bM,
                 const float* sM, const float* tM, const float* res,
                 float alpha, int act) {
  GemmP p;
  p.A = A; p.lda = lda; p.sA1 = sA1; p.sA2 = sA2;
  p.Bm = Bm; p.ldb = ldb; p.sB1 = sB1; p.sB2 = sB2;
  p.C = C; p.ldc = ldc; p.sC1 = sC1; p.sC2 = sC2;
  p.biasN = bN; p.biasM = bM; p.scaleM = sM; p.shiftM = tM; p.resid = res;
  p.alpha = alpha; p.act = act; p.M = M; p.N = N; p.K = K; p.bz2 = z2;
  if (N % 64 == 0) {
    dim3 grid((N + 255) / 256, M / 16, z1 * z2);
    if (btr) wmma_gemm<1, 4><<<grid, dim3(128), 0, st>>>(p);
    else     wmma_gemm<0, 4><<<grid, dim3(128), 0, st>>>(p);
  } else {
    dim3 grid((N + 63) / 64, M / 16, z1 * z2);
    if (btr) wmma_gemm<1, 1><<<grid, dim3(128), 0, st>>>(p);
    else     wmma_gemm<0, 1><<<grid, dim3(128), 0, st>>>(p);
  }
}

// workspace layout (float offsets)
#define F_BTC 65536L
static const long O_VT = 0;
static const long O_AT = O_VT + F_BTC;
static const long O_QP = O_AT + F_BTC;
static const long O_KP = O_QP + F_BTC;
static const long O_VP = O_KP + F_BTC;
static const long O_ATT = O_VP + F_BTC;               // 524288
static const long O_OT = O_ATT + 524288L;
static const long O_VGB = O_OT + F_BTC;
static const long O_GV = O_VGB + F_BTC;
static const long O_X = O_GV + F_BTC;
static const long O_Y0 = O_X + F_BTC;                 // 5242880
#define F_BIG 2621440L
static const long O_B0 = O_Y0 + 5242880L;
static const long O_B1 = O_B0 + F_BIG;
static const long O_B2 = O_B1 + F_BIG;
static const long O_B3 = O_B2 + F_BIG;
static const long O_B4 = O_B3 + F_BIG;
static const long O_B5 = O_B4 + F_BIG;
static const long O_B6 = O_B5 + F_BIG;
static const long O_B7 = O_B6 + F_BIG;
static const long O_PA = O_B7 + F_BIG;                // pooled g  (655360)
static const long O_PB = O_PA + 655360L;              // pooled phi
static const long O_FC = O_PB + 655360L;              // chan f (32768)
static const long O_S1 = O_FC + 32768L;
static const long O_S2 = O_S1 + 20480L;
static const long O_S3 = O_S2 + 20480L;

extern "C" void kernel_launch(void* const* d_in, const int* in_sizes, int n_in,
                              void* d_out, int out_size, void* d_ws,
                              size_t ws_size, hipStream_t stream) {
  (void)in_sizes; (void)n_in; (void)out_size; (void)ws_size;
  auto IN = [&](int i) { return (const float*)d_in[i]; };
  float* ws = (float*)d_ws;
  // input indices (setup_inputs() dict order, recursive):
  // 0 v_feat, 1 a_feat, 2 smp_weight,
  // cmg: 3 v2a_in_w, 4 v2a_in_b, 5 v2a_out_w, 6 v2a_out_b,
  //      7 a2v_in_w, 8 a2v_in_b, 9 a2v_out_w, 10 a2v_out_b,
  //      11 gate_v_w, 12 gate_v_b, 13 gate_a_w, 14 gate_a_b,
  // 15 w0, 16 b0, 17 w1a, 18 b1a, 19 w1b, 20 b1b,
  // prb: conv0_0 21-24 (w,b,s,t), conv0_1 25-28, conv1 29-32, conv4 33-36,
  //      conv2 37-38, conv3 39-40, conv5 41-42,
  //      p: g 43-46, theta 47-50, phi 51-54, W 55-58,
  //      c: g 59-62, theta 63-66, phi 67-70, W 71-74
  const int n_btc = NB * CINC * TSEQ;
  const int eb = (n_btc + 255) / 256;

  // transposes  [B,C,T] -> [B,T,C]
  transpose_ct<<<eb, 256, 0, stream>>>(IN(0), ws + O_VT);
  transpose_ct<<<eb, 256, 0, stream>>>(IN(1), ws + O_AT);

  // ---- MHA a2v: q from v, k/v from a (v2a branch is dead code; skipped) ----
  const float scale = 0.17677669529663689f;  // 1/sqrt(32)
  gemm(stream, ws + O_VT, CINC, (long)TSEQ * CINC, 0, IN(7), CINC, 0, 0, 1,
       ws + O_QP, CINC, (long)TSEQ * CINC, 0, TSEQ, CINC, CINC, NB, 1, IN(8),
       nullptr, nullptr, nullptr, nullptr, scale, 0);
  gemm(stream, ws + O_AT, CINC, (long)TSEQ * CINC, 0, IN(7) + CINC * CINC,
       CINC, 0, 0, 1, ws + O_KP, CINC, (long)TSEQ * CINC, 0, TSEQ, CINC, CINC,
       NB, 1, IN(8) + CINC, nullptr, nullptr, nullptr, nullptr, 1.f, 0);
  gemm(stream, ws + O_AT, CINC, (long)TSEQ * CINC, 0, IN(7) + 2 * CINC * CINC,
       CINC, 0, 0, 1, ws + O_VP, CINC, (long)TSEQ * CINC, 0, TSEQ, CINC, CINC,
       NB, 1, IN(8) + 2 * CINC, nullptr, nullptr, nullptr, nullptr, 1.f, 0);
  // scores[b,h,q,k]
  gemm(stream, ws + O_QP, CINC, (long)TSEQ * CINC, DHD, ws + O_KP, CINC,
       (long)TSEQ * CINC, DHD, 1, ws + O_ATT, TSEQ,
       (long)NHEADS * TSEQ * TSEQ, (long)TSEQ * TSEQ, TSEQ, TSEQ, DHD, NB,
       NHEADS, nullptr, nullptr, nullptr, nullptr, nullptr, 1.f, 0);
  softmax_rows<<<NB * NHEADS * TSEQ, 256, 0, stream>>>(ws + O_ATT, TSEQ);
  // o = att @ vp   (N = 32 -> NT1 path)
  gemm(stream, ws + O_ATT, TSEQ, (long)NHEADS * TSEQ * TSEQ,
       (long)TSEQ * TSEQ, ws + O_VP, CINC, (long)TSEQ * CINC, DHD, 0,
       ws + O_OT, CINC, (long)TSEQ * CINC, DHD, TSEQ, DHD, TSEQ, NB, NHEADS,
       nullptr, nullptr, nullptr, nullptr, nullptr, 1.f, 0);
  // out proj -> v_g
  gemm(stream, ws + O_OT, CINC, (long)TSEQ * CINC, 0, IN(9), CINC, 0, 0, 1,
       ws + O_VGB, CINC, (long)TSEQ * CINC, 0, TSEQ, CINC, CINC, NB, 1,
       IN(10), nullptr, nullptr, nullptr, nullptr, 1.f, 0);
  // gate: gv = sigmoid(vT @ W^T + b)
  gemm(stream, ws + O_VT, CINC, (long)TSEQ * CINC, 0, IN(11), CINC, 0, 0, 1,
       ws + O_GV, CINC, (long)TSEQ * CINC, 0, TSEQ, CINC, CINC, NB, 1, IN(12),
       nullptr, nullptr, nullptr, nullptr, 1.f, 3);
  // fuse + transpose back -> x [B,C,T]
  gate_fuse<<<eb, 256, 0, stream>>>(ws + O_GV, ws + O_VGB, ws + O_VT,
                                    ws + O_X);

  // ---- fused BM layer + block0 -> y0 [B,256,40,256] ----
  bm_block0<<<dim3(TSEQ / 16, NDUR, NB), 256, 128 * 160 * 4, stream>>>(
      ws + O_X, IN(2), IN(15), IN(16), ws + O_Y0);

  // ---- block1: 1x1 conv (leaky) then 3x3 conv (leaky) ----
  gemm(stream, IN(17), D0, 0, 0, ws + O_Y0, CHW, (long)D0 * CHW, 0, 0,
       ws + O_B0, CHW, (long)D1 * CHW, 0, D1, CHW, D0, NB, 1, nullptr, IN(18),
       nullptr, nullptr, nullptr, 1.f, 1);
  {
    const long tot = (long)NB * D1 * CHW;
    conv3x3<<<(tot + 255) / 256, 256, 0, stream>>>(ws + O_B0, IN(19), IN(20),
                                                   nullptr, nullptr, ws + O_B1,
                                                   D1, D1, 1);
  }
  const long tot1 = (long)NB * D1 * CHW;
  const int cg = (tot1 + 255) / 256;

  // ---- PRB: conv0_0 / conv0_1 (conv+BN+relu) ----
  conv3x3<<<cg, 256, 0, stream>>>(ws + O_B1, IN(21), IN(22), IN(23), IN(24),
                                  ws + O_B2, D1, D1, 2);  // t0
  conv3x3<<<cg, 256, 0, stream>>>(ws + O_B1, IN(25), IN(26), IN(27), IN(28),
                                  ws + O_B3, D1, D1, 2);  // t1

  // ---- positional attention branch (input/residual t0=B2) ----
  gemm(stream, IN(43), D1, 0, 0, ws + O_B2, CHW, (long)D1 * CHW, 0, 0,
       ws + O_B0, CHW, (long)D1 * CHW, 0, D1, CHW, D1, NB, 1, nullptr, IN(44),
       IN(45), IN(46), nullptr, 1.f, 2);
  pool2k<<<(NB * D1 * PHW + 255) / 256, 256, 0, stream>>>(ws + O_B0,
                                                          ws + O_PA);
  gemm(stream, IN(51), D1, 0, 0, ws + O_B2, CHW, (long)D1 * CHW, 0, 0,
       ws + O_B0, CHW, (long)D1 * CHW, 0, D1, CHW, D1, NB, 1, nullptr, IN(52),
       IN(53), IN(54), nullptr, 1.f, 2);
  pool2k<<<(NB * D1 * PHW + 255) / 256, 256, 0, stream>>>(ws + O_B0,
                                                          ws + O_PB);
  gemm(stream, IN(47), D1, 0, 0, ws + O_B2, CHW, (long)D1 * CHW, 0, 0,
       ws + O_B1, CHW, (long)D1 * CHW, 0, D1, CHW, D1, NB, 1, nullptr, IN(48),
       IN(49), IN(50), nullptr, 1.f, 2);
  pos_attn<<<dim3(CHW / 16, 1, NB), 256, 16 * PHW * 4, stream>>>(
      ws + O_B1, ws + O_PB, ws + O_PA, ws + O_B4);
  // x_p = BN(W @ y_pos + b) + t0  -> B5
  gemm(stream, IN(55), D1, 0, 0, ws + O_B4, CHW, (long)D1 * CHW, 0, 0,
       ws + O_B5, CHW, (long)D1 * CHW, 0, D1, CHW, D1, NB, 1, nullptr, IN(56),
       IN(57), IN(58), ws + O_B2, 1.f, 0);

  // ---- channel attention branch (input/residual t1=B3) ----
  gemm(stream, IN(59), D1, 0, 0, ws + O_B3, CHW, (long)D1 * CHW, 0, 0,
       ws + O_B0, CHW, (long)D1 * CHW, 0, D1, CHW, D1, NB, 1, nullptr, IN(60),
       IN(61), IN(62), nullptr, 1.f, 2);  // g -> B0
  gemm(stream, IN(63), D1, 0, 0, ws + O_B3, CHW, (long)D1 * CHW, 0, 0,
       ws + O_B1, CHW, (long)D1 * CHW, 0, D1, CHW, D1, NB, 1, nullptr, IN(64),
       IN(65), IN(66), nullptr, 1.f, 2);  // theta -> B1
  gemm(stream, IN(67), D1, 0, 0, ws + O_B3, CHW, (long)D1 * CHW, 0, 0,
       ws + O_B4, CHW, (long)D1 * CHW, 0, D1, CHW, D1, NB, 1, nullptr, IN(68),
       IN(69), IN(70), nullptr, 1.f, 2);  // phi -> B4
  // f = th @ ph^T  [B,128,128]
  gemm(stream, ws + O_B1, CHW, (long)D1 * CHW, 0, ws + O_B4, CHW,
       (long)D1 * CHW, 0, 1, ws + O_FC, D1, (long)D1 * D1, 0, D1, D1, CHW, NB,
       1, nullptr, nullptr, nullptr, nullptr, nullptr, 1.f, 0);
  softmax_rows<<<NB * D1, 256, 0, stream>>>(ws + O_FC, D1);
  // y = f @ g -> B2 (t0 no longer needed)
  gemm(stream, ws + O_FC, D1, (long)D1 * D1, 0, ws + O_B0, CHW,
       (long)D1 * CHW, 0, 0, ws + O_B2, CHW, (long)D1 * CHW, 0, D1, CHW, D1,
       NB, 1, nullptr, nullptr, nullptr, nullptr, nullptr, 1.f, 0);
  chan_shuffle<<<cg, 256, 0, stream>>>(ws + O_B2, ws + O_B6);
  // x_c = BN(W @ y2 + b) + t1 -> B7
  gemm(stream, IN(71), D1, 0, 0, ws + O_B6, CHW, (long)D1 * CHW, 0, 0,
       ws + O_B7, CHW, (long)D1 * CHW, 0, D1, CHW, D1, NB, 1, nullptr, IN(72),
       IN(73), IN(74), ws + O_B3, 1.f, 0);

  // ---- heads ----
  conv3x3<<<cg, 256, 0, stream>>>(ws + O_B5, IN(29), IN(30), IN(31), IN(32),
                                  ws + O_B0, D1, D1, 2);  // x_p0
  conv3x3<<<cg, 256, 0, stream>>>(ws + O_B7, IN(33), IN(34), IN(35), IN(36),
                                  ws + O_B1, D1, D1, 2);  // x_c0
  const int c1g = (NB * CHW + 255) / 256;
  conv3x3<<<c1g, 256, 0, stream>>>(ws + O_B0, IN(37), IN(38), nullptr, nullptr,
                                   ws + O_S1, 1, D1, 3);  // x_p1
  conv3x3<<<c1g, 256, 0, stream>>>(ws + O_B1, IN(41), IN(42), nullptr, nullptr,
                                   ws + O_S2, 1, D1, 3);  // x_c1
  add2<<<cg, 256, 0, stream>>>(ws + O_B0, ws + O_B1, ws + O_B2,
                               (int)((long)NB * D1 * CHW));
  conv3x3<<<c1g, 256, 0, stream>>>(ws + O_B2, IN(39), IN(40), nullptr, nullptr,
                                   ws + O_S3, 1, D1, 3);  // x_pc
  combine3<<<c1g, 256, 0, stream>>>(ws + O_S1, ws + O_S2, ws + O_S3,
                                    (float*)d_out, NB * CHW);
}